// PointNet2SemSeg_10213432230207
// MI455X (gfx1250) — compile-verified
//
#include <hip/hip_runtime.h>
#include <cstdint>

// ---------------------------------------------------------------------------
// PointNet++ sem-seg for MI455X (gfx1250, wave32).
// - GEMMs: V_WMMA_F32_16X16X4_F32, one wave per 32x32 f32 tile (2x2 register
//   tiling: 4 accumulators, A/B fragment reuse -> 6 loads per 4 WMMAs).
//   Inputs zero-padded (K->mult of 4, weight N->mult of 32) so the inner loop
//   is branch-free.
// - Ball-query xyz tiles staged to LDS with GLOBAL_LOAD_ASYNC_TO_LDS_B32
//   (ASYNCcnt) instead of load->VGPR->ds_store.
// Workload is latency/gather bound (~20 GFLOP total, working set < L2), so
// f32 WMMA keeps reference numerics at no roofline cost.
// ---------------------------------------------------------------------------

typedef __attribute__((ext_vector_type(2))) float v2f;
typedef __attribute__((ext_vector_type(8))) float v8f;

#define EPSF 1e-5f
static const int BB = 8;
static const int NN = 8192;

// ------------------------- FPS (farthest point sampling) -------------------
__global__ void fps_kernel(const float* __restrict__ xyz, int* __restrict__ idx,
                           int Nn, int npoint) {
  const int T = 256;
  int b = blockIdx.x;
  int tid = threadIdx.x;
  const float* px = xyz + (size_t)b * Nn * 3;
  float dist[32];
  int nper = (Nn + T - 1) / T;
  for (int i = 0; i < nper; ++i) dist[i] = 1e10f;

  __shared__ float s_val[256];
  __shared__ int   s_idx[256];
  __shared__ float s_cent[3];
  __shared__ int   s_far;
  if (tid == 0) s_far = 0;
  __syncthreads();

  for (int it = 0; it < npoint; ++it) {
    int far = s_far;
    if (tid == 0) idx[(size_t)b * npoint + it] = far;
    if (tid < 3) s_cent[tid] = px[(size_t)far * 3 + tid];
    __syncthreads();
    float cx = s_cent[0], cy = s_cent[1], cz = s_cent[2];
    float best = -1.0f; int besti = 0;
    for (int i = 0; i < nper; ++i) {
      int n = tid + i * T;
      if (n < Nn) {
        float dx = px[n*3+0]-cx, dy = px[n*3+1]-cy, dz = px[n*3+2]-cz;
        float d = dx*dx + dy*dy + dz*dz;
        float dm = fminf(dist[i], d);
        dist[i] = dm;
        if (dm > best) { best = dm; besti = n; }
      }
    }
    s_val[tid] = best; s_idx[tid] = besti;
    __syncthreads();
    for (int s = 128; s > 0; s >>= 1) {
      if (tid < s) {
        if (s_val[tid+s] > s_val[tid]) { s_val[tid] = s_val[tid+s]; s_idx[tid] = s_idx[tid+s]; }
      }
      __syncthreads();
    }
    if (tid == 0) s_far = s_idx[0];
    __syncthreads();
  }
}

// ------------------------- gather sampled centroids ------------------------
__global__ void gather_xyz_kernel(const float* __restrict__ xyz, const int* __restrict__ idx,
                                  float* __restrict__ out, int Nn, int S) {
  int t = blockIdx.x * blockDim.x + threadIdx.x;
  if (t >= BB * S) return;
  int b = t / S;
  int n = idx[t];
  const float* p = xyz + ((size_t)b * Nn + n) * 3;
  out[t*3+0] = p[0]; out[t*3+1] = p[1]; out[t*3+2] = p[2];
}

// ------------------------- ball query (async LDS staging) ------------------
#define BQ_TILE 1024
__global__ void ball_query_kernel(const float* __restrict__ xyz, const float* __restrict__ new_xyz,
                                  int* __restrict__ gidx, int Nn, int S, float r2, int nsample) {
  __shared__ float sx[BQ_TILE * 3];
  int qblocks = (S + 255) / 256;
  int b = blockIdx.x / qblocks;
  int s = (blockIdx.x % qblocks) * 256 + threadIdx.x;
  bool valid = s < S;
  float qx = 0.f, qy = 0.f, qz = 0.f;
  size_t out_base = 0;
  if (valid) {
    const float* q = new_xyz + ((size_t)b * S + s) * 3;
    qx = q[0]; qy = q[1]; qz = q[2];
    out_base = ((size_t)b * S + s) * nsample;
  }
  const float* px = xyz + (size_t)b * Nn * 3;
  int cnt = 0;
  for (int base = 0; base < Nn; base += BQ_TILE) {
    int tlen = (Nn - base < BQ_TILE) ? (Nn - base) : BQ_TILE;
    __syncthreads();
    // async global -> LDS copy of this xyz tile (ASYNCcnt-tracked DMA path)
    for (int i = threadIdx.x; i < tlen * 3; i += 256) {
      uint32_t lds = (uint32_t)(uintptr_t)(&sx[i]);      // LDS byte offset (flat LDS addr truncates)
      uint64_t ga  = (uint64_t)(uintptr_t)(px + (size_t)base * 3 + i);
      asm volatile("global_load_async_to_lds_b32 %0, %1, off"
                   :: "v"(lds), "v"(ga) : "memory");
    }
    asm volatile("s_wait_asynccnt 0x0" ::: "memory");
    __syncthreads();
    if (valid && cnt < nsample) {
      for (int j = 0; j < tlen; ++j) {
        float dx = sx[j*3+0]-qx, dy = sx[j*3+1]-qy, dz = sx[j*3+2]-qz;
        float d = dx*dx + dy*dy + dz*dz;
        if (d <= r2) {
          gidx[out_base + cnt] = base + j;
          if (++cnt == nsample) break;
        }
      }
    }
  }
  if (valid) {
    int first = (cnt > 0) ? gidx[out_base] : 0;
    for (int k = cnt; k < nsample; ++k) gidx[out_base + k] = first;
  }
}

// ------------------------- grouping (writes K-padded rows) -----------------
__global__ void group_kernel(const float* __restrict__ xyz, const float* __restrict__ pts,
                             const float* __restrict__ new_xyz, const int* __restrict__ gidx,
                             float* __restrict__ out, int Nn, int S, int nsample, int Cp, int Kp) {
  int t = blockIdx.x * blockDim.x + threadIdx.x;
  int total = BB * S * nsample;
  if (t >= total) return;
  int bs = t / nsample;
  int b = bs / S;
  int n = gidx[t];
  const float* pz = xyz + ((size_t)b * Nn + n) * 3;
  const float* pc = new_xyz + (size_t)bs * 3;
  float* o = out + (size_t)t * Kp;
  o[0] = pz[0] - pc[0]; o[1] = pz[1] - pc[1]; o[2] = pz[2] - pc[2];
  if (pts) {
    const float* pp = pts + ((size_t)b * Nn + n) * Cp;
    for (int c = 0; c < Cp; ++c) o[3 + c] = pp[c];
  }
  for (int c = 3 + Cp; c < Kp; ++c) o[c] = 0.f;   // zero pad -> branch-free GEMM
}

// ------------------------- weight repack: zero-padded [Kp x Ncp] -----------
__global__ void pad_w_kernel(const float* __restrict__ W, float* __restrict__ Wp,
                             int K, int Nc, int Kp, int Ncp) {
  int t = blockIdx.x * blockDim.x + threadIdx.x;
  if (t >= Kp * Ncp) return;
  int k = t / Ncp, n = t % Ncp;
  Wp[t] = (k < K && n < Nc) ? W[(size_t)k * Nc + n] : 0.f;
}

// ------------------------- WMMA f32 GEMM (32x32 tile / wave) ---------------
// Y[M x Nc] = X[M x Kp] @ Wp[Kp x Ncp] (+bias). One wave per 32x32 tile,
// 2x2 register tiling: 4 v8f accumulators, A/B fragment reuse.
// Kp % 4 == 0, Ncp % 32 == 0, M % 32 == 0 guaranteed by producers.
// Y has row stride Nc (real); stores guarded by n < Nc.
__global__ __launch_bounds__(32)
void gemm_wmma_kernel(const float* __restrict__ X, const float* __restrict__ Wp,
                      const float* __restrict__ bias, float* __restrict__ Y,
                      int M, int Kp, int Ncp, int Nc) {
  int ntN = Ncp >> 5;
  int tile = blockIdx.x;
  int tm = (tile / ntN) * 32;
  int tn = (tile % ntN) * 32;
  int lane = threadIdx.x;
  int half = lane >> 4;
  int l16  = lane & 15;
  int n0 = tn + l16;
  int n1 = tn + 16 + l16;

  v8f acc00 = {0.f,0.f,0.f,0.f,0.f,0.f,0.f,0.f};   // (tm..tm+15)   x (tn..tn+15)
  v8f acc01 = acc00;                               // (tm..tm+15)   x (tn+16..)
  v8f acc10 = acc00;                               // (tm+16..)     x (tn..tn+15)
  v8f acc11 = acc00;                               // (tm+16..)     x (tn+16..)

  const float* Xrow0 = X + (size_t)(tm + l16) * Kp;
  const float* Xrow1 = Xrow0 + (size_t)16 * Kp;
  const v2f*   Ar0   = (const v2f*)Xrow0;                   // pair (kb/2 + half)
  const v2f*   Ar1   = (const v2f*)Xrow1;
  const float* Wc0   = Wp + (size_t)(2 * half) * Ncp + n0;  // W[kb+2*half][n0]
  const float* Wc1   = Wp + (size_t)(2 * half) * Ncp + n1;

#pragma unroll 2
  for (int kb = 0; kb < Kp; kb += 4) {
    __builtin_prefetch(Xrow0 + kb + 64, 0, 3);              // global_prefetch_b8
    int pi = (kb >> 1) + half;
    v2f a0 = Ar0[pi];
    v2f a1 = Ar1[pi];
    v2f b0, b1;
    b0.x = Wc0[(size_t)kb * Ncp];
    b0.y = Wc0[(size_t)(kb + 1) * Ncp];
    b1.x = Wc1[(size_t)kb * Ncp];
    b1.y = Wc1[(size_t)(kb + 1) * Ncp];
    acc00 = __builtin_amdgcn_wmma_f32_16x16x4_f32(false, a0, false, b0, (short)0, acc00, false, false);
    acc01 = __builtin_amdgcn_wmma_f32_16x16x4_f32(false, a0, false, b1, (short)0, acc01, false, false);
    acc10 = __builtin_amdgcn_wmma_f32_16x16x4_f32(false, a1, false, b0, (short)0, acc10, false, false);
    acc11 = __builtin_amdgcn_wmma_f32_16x16x4_f32(false, a1, false, b1, (short)0, acc11, false, false);
  }

  if (n0 < Nc) {
    float bv = bias ? bias[n0] : 0.f;
#pragma unroll
    for (int v = 0; v < 8; ++v) {
      int m = tm + v + 8 * half;
      Y[(size_t)m * Nc + n0]        = acc00[v] + bv;
      Y[(size_t)(m + 16) * Nc + n0] = acc10[v] + bv;
    }
  }
  if (n1 < Nc) {
    float bv = bias ? bias[n1] : 0.f;
#pragma unroll
    for (int v = 0; v < 8; ++v) {
      int m = tm + v + 8 * half;
      Y[(size_t)m * Nc + n1]        = acc01[v] + bv;
      Y[(size_t)(m + 16) * Nc + n1] = acc11[v] + bv;
    }
  }
}

// ------------------------- BatchNorm (deterministic 2-level reduce) --------
__global__ void bn_partial_kernel(const float* __restrict__ Y, float* __restrict__ part,
                                  int M, int Nc, int nchunks) {
  int t = blockIdx.x * blockDim.x + threadIdx.x;
  if (t >= nchunks * Nc) return;
  int c = t % Nc, chunk = t / Nc;
  int rows = (M + nchunks - 1) / nchunks;
  int r0 = chunk * rows;
  int r1 = (r0 + rows < M) ? (r0 + rows) : M;
  float s = 0.f, s2 = 0.f;
  for (int m = r0; m < r1; ++m) {
    float v = Y[(size_t)m * Nc + c];
    s += v; s2 += v * v;
  }
  part[((size_t)chunk * 2 + 0) * Nc + c] = s;
  part[((size_t)chunk * 2 + 1) * Nc + c] = s2;
}

__global__ void bn_finalize_kernel(const float* __restrict__ part, float* __restrict__ stats,
                                   int Nc, int nchunks) {
  int c = blockIdx.x * blockDim.x + threadIdx.x;
  if (c >= Nc) return;
  float s = 0.f, s2 = 0.f;
  for (int k = 0; k < nchunks; ++k) {
    s  += part[((size_t)k * 2 + 0) * Nc + c];
    s2 += part[((size_t)k * 2 + 1) * Nc + c];
  }
  stats[c] = s; stats[Nc + c] = s2;
}

__global__ void bn_relu_kernel(float* __restrict__ Y, const float* __restrict__ stats,
                               const float* __restrict__ g, const float* __restrict__ bta,
                               int M, int Nc) {
  size_t t = (size_t)blockIdx.x * blockDim.x + threadIdx.x;
  if (t >= (size_t)M * Nc) return;
  int c = (int)(t % Nc);
  float invM = 1.f / (float)M;
  float mean = stats[c] * invM;
  float var  = stats[Nc + c] * invM - mean * mean;
  float v = (Y[t] - mean) * rsqrtf(var + EPSF) * g[c] + bta[c];
  Y[t] = v > 0.f ? v : 0.f;
}

// ------------------------- max pool over nsample ---------------------------
__global__ void maxpool_kernel(const float* __restrict__ Y, float* __restrict__ out,
                               int BS, int ns, int C) {
  int t = blockIdx.x * blockDim.x + threadIdx.x;
  if (t >= BS * C) return;
  int bs = t / C, c = t % C;
  const float* p = Y + (size_t)bs * ns * C + c;
  float m = p[0];
  for (int j = 1; j < ns; ++j) m = fmaxf(m, p[(size_t)j * C]);
  out[t] = m;
}

// ------------------------- 3-NN for feature propagation --------------------
__global__ void three_nn_kernel(const float* __restrict__ xyz1, const float* __restrict__ xyz2,
                                int* __restrict__ idx3, float* __restrict__ w3,
                                int N1, int S2) {
  int t = blockIdx.x * blockDim.x + threadIdx.x;
  if (t >= BB * N1) return;
  int b = t / N1;
  const float* p = xyz1 + (size_t)t * 3;
  float x = p[0], y = p[1], z = p[2];
  const float* q = xyz2 + (size_t)b * S2 * 3;
  float d0 = 1e30f, d1 = 1e30f, d2 = 1e30f;
  int i0 = 0, i1 = 0, i2 = 0;
  for (int s = 0; s < S2; ++s) {
    float dx = q[s*3+0]-x, dy = q[s*3+1]-y, dz = q[s*3+2]-z;
    float d = dx*dx + dy*dy + dz*dz;
    if (d < d0)      { d2=d1;i2=i1; d1=d0;i1=i0; d0=d;i0=s; }
    else if (d < d1) { d2=d1;i2=i1; d1=d;i1=s; }
    else if (d < d2) { d2=d;i2=s; }
  }
  float w0 = 1.f/(d0+1e-8f), w1 = 1.f/(d1+1e-8f), w2 = 1.f/(d2+1e-8f);
  float ws = w0 + w1 + w2;
  idx3[(size_t)t*3+0] = i0; idx3[(size_t)t*3+1] = i1; idx3[(size_t)t*3+2] = i2;
  w3[(size_t)t*3+0] = w0/ws; w3[(size_t)t*3+1] = w1/ws; w3[(size_t)t*3+2] = w2/ws;
}

// out[b,n, :C1] = points1, out[b,n, C1:] = sum_k w_k * points2[idx_k]
__global__ void fp_interp_concat_kernel(const float* __restrict__ points1,
                                        const float* __restrict__ points2,
                                        const int* __restrict__ idx3, const float* __restrict__ w3,
                                        float* __restrict__ out,
                                        int N1, int S2, int C1, int C2) {
  int C = C1 + C2;
  size_t t = (size_t)blockIdx.x * blockDim.x + threadIdx.x;
  size_t total = (size_t)BB * N1 * C;
  if (t >= total) return;
  int c = (int)(t % C);
  size_t bn = t / C;
  int b = (int)(bn / N1);
  if (c < C1) {
    out[t] = points1[bn * C1 + c];
  } else {
    int cc = c - C1;
    const int*   ii = idx3 + bn * 3;
    const float* ww = w3  + bn * 3;
    const float* p2 = points2 + (size_t)b * S2 * C2;
    out[t] = ww[0] * p2[(size_t)ii[0] * C2 + cc]
           + ww[1] * p2[(size_t)ii[1] * C2 + cc]
           + ww[2] * p2[(size_t)ii[2] * C2 + cc];
  }
}

// ===========================================================================
// Host orchestration
// ===========================================================================
struct Lyr { const float *W, *g, *b; };

extern "C" void kernel_launch(void* const* d_in, const int* in_sizes, int n_in,
                              void* d_out, int out_size, void* d_ws, size_t ws_size,
                              hipStream_t stream) {
  (void)n_in; (void)out_size; (void)ws_size;

  // ---- parameter mapping ------------------------------------------------
  Lyr sa1L[3], sa2L[3], sa3L[3], sa4L[3], fp4L[2], fp3L[2], fp2L[2], fp1L[3];
  const float *clsW1, *clsg1, *clsb1, *clsW2, *clsb2;
  const float *xyz, *feat;
  auto F = [&](int i) { return (const float*)d_in[i]; };

  bool insertion = (in_sizes[0] == BB * NN * 3);  // xyz first => insertion-order flatten
  if (!insertion) {
    // jax tree flatten: sorted dict keys at every level
    feat  = F(0);
    clsW1 = F(1); clsW2 = F(2); clsb1 = F(3); clsb2 = F(4); clsg1 = F(5);
    int i = 6;
    auto rd = [&](Lyr* L, int nl) {
      for (int j = 0; j < nl; ++j) { L[j].W = F(i); L[j].b = F(i+1); L[j].g = F(i+2); i += 3; }
    };
    rd(fp1L,3); rd(fp2L,2); rd(fp3L,2); rd(fp4L,2);
    rd(sa1L,3); rd(sa2L,3); rd(sa3L,3); rd(sa4L,3);
    xyz = F(i);
  } else {
    xyz = F(0); feat = F(1);
    int i = 2;
    auto rd = [&](Lyr* L, int nl) {
      for (int j = 0; j < nl; ++j) { L[j].W = F(i); L[j].g = F(i+1); L[j].b = F(i+2); i += 3; }
    };
    rd(sa1L,3); rd(sa2L,3); rd(sa3L,3); rd(sa4L,3);
    rd(fp4L,2); rd(fp3L,2); rd(fp2L,2); rd(fp1L,3);
    clsW1 = F(i++); clsg1 = F(i++); clsb1 = F(i++); clsW2 = F(i++); clsb2 = F(i++);
  }

  // ---- workspace carve-up ------------------------------------------------
  uint8_t* ws = (uint8_t*)d_ws;
  size_t off = 0;
  auto alloc_f = [&](size_t n) -> float* {
    float* p = (float*)(ws + off); off += ((n * 4 + 255) & ~(size_t)255); return p;
  };
  auto alloc_i = [&](size_t n) -> int* {
    int* p = (int*)(ws + off); off += ((n * 4 + 255) & ~(size_t)255); return p;
  };

  const size_t ARENA = (size_t)17 * 1024 * 1024;  // floats; fits sa1 64ch output
  float* arenaA = alloc_f(ARENA);
  float* arenaB = alloc_f(ARENA);
  float* stats  = alloc_f(1024);
  float* part   = alloc_f((size_t)128 * 2 * 512);
  float* wpad   = alloc_f((size_t)768 * 512);     // padded weight scratch
  int*   idxbuf = alloc_i(BB * 1024);
  int*   gidxb  = alloc_i((size_t)BB * 1024 * 32);
  int*   idx3b  = alloc_i((size_t)BB * NN * 3);
  float* w3b    = alloc_f((size_t)BB * NN * 3);
  float* l1x = alloc_f(BB*1024*3); float* l2x = alloc_f(BB*256*3);
  float* l3x = alloc_f(BB*64*3);   float* l4x = alloc_f(BB*16*3);
  float* l1p = alloc_f((size_t)BB*1024*64);  float* l2p = alloc_f((size_t)BB*256*128);
  float* l3p = alloc_f((size_t)BB*64*256);   float* l4p = alloc_f((size_t)BB*16*512);
  float* l3f = alloc_f((size_t)BB*64*256);   float* l2f = alloc_f((size_t)BB*256*256);
  float* l1f = alloc_f((size_t)BB*1024*128); float* l0f = alloc_f((size_t)BB*NN*128);

  const int NCH = 128;  // bn reduction chunks

  // ---- GEMM (+optional bias) with padded weights -------------------------
  auto gemm = [&](const float* X, float* Y, const float* W, const float* bias,
                  int M, int K, int Kp, int Nc) {
    int Ncp = (Nc + 31) & ~31;                       // 32x32 tiles per wave
    pad_w_kernel<<<(Kp * Ncp + 255) / 256, 256, 0, stream>>>(W, wpad, K, Nc, Kp, Ncp);
    int ntM = M / 32, ntN = Ncp / 32;
    gemm_wmma_kernel<<<ntM * ntN, 32, 0, stream>>>(X, wpad, bias, Y, M, Kp, Ncp, Nc);
  };

  // ---- conv + bn + relu (X already Kp-padded) ----------------------------
  auto conv_bn_relu = [&](const float* X, float* Y, const Lyr& L, int M, int K, int Kp, int Nc) {
    gemm(X, Y, L.W, nullptr, M, K, Kp, Nc);
    bn_partial_kernel<<<(NCH * Nc + 255) / 256, 256, 0, stream>>>(Y, part, M, Nc, NCH);
    bn_finalize_kernel<<<(Nc + 255) / 256, 256, 0, stream>>>(part, stats, Nc, NCH);
    size_t tot = (size_t)M * Nc;
    bn_relu_kernel<<<(unsigned)((tot + 255) / 256), 256, 0, stream>>>(Y, stats, L.g, L.b, M, Nc);
  };

  // ---- set-abstraction layer --------------------------------------------
  auto run_sa = [&](const float* xin, int Nin, const float* pin, int Cp,
                    int S, float radius, const Lyr* mlp, const int* ch,
                    float* nxyz, float* outp) {
    fps_kernel<<<BB, 256, 0, stream>>>(xin, idxbuf, Nin, S);
    gather_xyz_kernel<<<(BB * S + 255) / 256, 256, 0, stream>>>(xin, idxbuf, nxyz, Nin, S);
    int qb = (S + 255) / 256;
    ball_query_kernel<<<BB * qb, 256, 0, stream>>>(xin, nxyz, gidxb, Nin, S, radius * radius, 32);
    int M = BB * S * 32;
    int K0 = ch[0], Kp0 = (K0 + 3) & ~3;
    group_kernel<<<(M + 255) / 256, 256, 0, stream>>>(xin, pin, nxyz, gidxb, arenaA, Nin, S, 32, Cp, Kp0);
    conv_bn_relu(arenaA, arenaB, mlp[0], M, K0, Kp0, ch[1]);
    conv_bn_relu(arenaB, arenaA, mlp[1], M, ch[1], ch[1], ch[2]);
    conv_bn_relu(arenaA, arenaB, mlp[2], M, ch[2], ch[2], ch[3]);
    maxpool_kernel<<<(BB * S * ch[3] + 255) / 256, 256, 0, stream>>>(arenaB, outp, BB * S, 32, ch[3]);
  };

  // ---- feature-propagation layer (all K already mult of 4) ---------------
  auto run_fp = [&](const float* x1, int N1, const float* x2, int S2,
                    const float* p1, int C1, const float* p2, int C2,
                    const Lyr* mlp, const int* ch, int nl, float* outp) {
    three_nn_kernel<<<(BB * N1 + 255) / 256, 256, 0, stream>>>(x1, x2, idx3b, w3b, N1, S2);
    int C = C1 + C2;
    size_t tot = (size_t)BB * N1 * C;
    fp_interp_concat_kernel<<<(unsigned)((tot + 255) / 256), 256, 0, stream>>>(
        p1, p2, idx3b, w3b, arenaA, N1, S2, C1, C2);
    float* cur = arenaA; float* alt = arenaB;
    int M = BB * N1, K = C;
    for (int i = 0; i < nl; ++i) {
      float* dst = (i == nl - 1) ? outp : alt;
      conv_bn_relu(cur, dst, mlp[i], M, K, K, ch[i + 1]);
      K = ch[i + 1];
      float* t = cur; cur = dst; alt = t;
    }
  };

  // ---- full pipeline -----------------------------------------------------
  const int sa1c[4] = {9, 32, 32, 64};
  const int sa2c[4] = {67, 64, 64, 128};
  const int sa3c[4] = {131, 128, 128, 256};
  const int sa4c[4] = {259, 256, 256, 512};
  const int fp4c[3] = {768, 256, 256};
  const int fp3c[3] = {384, 256, 256};
  const int fp2c[3] = {320, 256, 128};
  const int fp1c[4] = {128, 128, 128, 128};

  run_sa(xyz, NN,   feat, 6,   1024, 0.1f, sa1L, sa1c, l1x, l1p);
  run_sa(l1x, 1024, l1p, 64,   256,  0.2f, sa2L, sa2c, l2x, l2p);
  run_sa(l2x, 256,  l2p, 128,  64,   0.4f, sa3L, sa3c, l3x, l3p);
  run_sa(l3x, 64,   l3p, 256,  16,   0.8f, sa4L, sa4c, l4x, l4p);

  run_fp(l3x, 64,   l4x, 16,   l3p, 256, l4p, 512, fp4L, fp4c, 2, l3f);
  run_fp(l2x, 256,  l3x, 64,   l2p, 128, l3f, 256, fp3L, fp3c, 2, l2f);
  run_fp(l1x, 1024, l2x, 256,  l1p, 64,  l2f, 256, fp2L, fp2c, 2, l1f);
  run_fp(xyz, NN,   l1x, 1024, nullptr, 0, l1f, 128, fp1L, fp1c, 3, l0f);

  // classifier head: conv_bn_relu(128->128) then logits = y @ W2 + b2
  Lyr clsL; clsL.W = clsW1; clsL.g = clsg1; clsL.b = clsb1;
  int Mc = BB * NN;
  conv_bn_relu(l0f, arenaA, clsL, Mc, 128, 128, 128);
  gemm(arenaA, (float*)d_out, clsW2, clsb2, Mc, 128, 128, 13);
}